// SuperpixelPooling_43404939494026
// MI455X (gfx1250) — compile-verified
//
#include <hip/hip_runtime.h>
#include <hip/hip_bf16.h>
#include <stdint.h>

// Problem constants (match reference).
constexpr int Bq  = 4;
constexpr int Cq  = 64;
constexpr int Hq  = 512;
constexpr int Wq  = 512;
constexpr int HW  = Hq * Wq;          // 262144 pixels / image
constexpr int NL  = 1000;             // labels
constexpr int NS  = 2048;             // edges / image

// Scatter-reduce tiling.
constexpr int BSZ    = 512;           // threads per block (16 wave32)
constexpr int CBS    = 16;            // channels per channel-block
constexpr int CB     = Cq / CBS;      // 4 channel blocks
constexpr int SLABS  = 16;            // pixel slabs per image
constexpr int PSLAB  = HW / SLABS;    // 16384 pixels per slab (64 KB of labels)

// Accumulate-kernel LDS: lsum[CBS*NL] f32 | lcnt[NL] f32 | lbuf[PSLAB] i32
constexpr int    LDS_FLOATS    = CBS * NL + NL;                 // 17000 floats = 68000 B (16B aligned)
constexpr size_t ACC_LDS_BYTES = (size_t)LDS_FLOATS * 4 + (size_t)PSLAB * 4; // 133536 B

// Gather tiling.
constexpr int ET = 256;               // edges per block
constexpr size_t GAT_LDS_BYTES = (size_t)ET * 2 * 4;            // 2048 B of staged edges

typedef unsigned int u32x4 __attribute__((ext_vector_type(4)));
typedef unsigned int u32x8 __attribute__((ext_vector_type(8)));

__device__ __forceinline__ void lds_atomic_add(float* p, float v) {
    __hip_atomic_fetch_add(p, v, __ATOMIC_RELAXED, __HIP_MEMORY_SCOPE_WORKGROUP);
}
__device__ __forceinline__ void glb_atomic_add(float* p, float v) {
    __hip_atomic_fetch_add(p, v, __ATOMIC_RELAXED, __HIP_MEMORY_SCOPE_AGENT);
}

// ---- CDNA5 async paths ----------------------------------------------------

// Per-lane async global -> LDS copy of 8 bytes (ASYNCcnt).
__device__ __forceinline__ void async_copy_b64(const void* gsrc, void* lds_dst) {
    unsigned           loff = (unsigned)(uintptr_t)lds_dst;   // low 32 bits = LDS offset
    unsigned long long gadr = (unsigned long long)(uintptr_t)gsrc;
    asm volatile("global_load_async_to_lds_b64 %0, %1, off"
                 :: "v"(loff), "v"(gadr) : "memory");
}
__device__ __forceinline__ void wait_asynccnt0() {
    asm volatile("s_wait_asynccnt 0x0" ::: "memory");
}
__device__ __forceinline__ void wait_tensorcnt0() {
#if __has_builtin(__builtin_amdgcn_s_wait_tensorcnt)
    __builtin_amdgcn_s_wait_tensorcnt(0);
#else
    asm volatile("s_wait_tensorcnt 0x0" ::: "memory");
#endif
}

// TDM: DMA `n_dwords` contiguous dwords from global memory into LDS.
// D# layout per CDNA5 ISA 8.3/8.4:
//   group0: [1:0]=count=1, [63:32]=lds_addr, [120:64]=global_addr, [127:126]=type=2
//   group1: [17:16]=data_size=2 (4B), tensor_dim0=[79:48], tile_dim0=[127:112],
//           tensor_dim0_stride=[207:160]; 1-D tile (tile_dim1=1, dims/strides above unused)
__device__ __forceinline__ void tdm_load_to_lds(const void* gsrc, void* lds_dst,
                                                unsigned n_dwords) {
    unsigned long long ga = (unsigned long long)(uintptr_t)gsrc;
    unsigned           la = (unsigned)(uintptr_t)lds_dst;
    u32x4 g0;
    g0.x = 1u;                                            // count=1, no gather, user D#
    g0.y = la;                                            // lds_addr
    g0.z = (unsigned)(ga & 0xFFFFFFFFu);                  // global_addr[31:0]
    g0.w = (unsigned)((ga >> 32) & 0x01FFFFFFu) | (2u << 30); // global_addr[56:32] | type=2
    u32x8 g1;
    g1.s0 = (2u << 16);                                   // data_size = 4 bytes
    g1.s1 = (n_dwords & 0xFFFFu) << 16;                   // tensor_dim0[15:0]
    g1.s2 = ((n_dwords >> 16) & 0xFFFFu) | (1u << 16);    // tensor_dim0[31:16] | tensor_dim1=1
    g1.s3 = (n_dwords & 0xFFFFu) << 16;                   // tile_dim0 (16-bit)
    g1.s4 = 1u;                                           // tile_dim1 = 1
    g1.s5 = n_dwords;                                     // tensor_dim0_stride[31:0]
    g1.s6 = 0u;                                           // stride hi / dim1_stride lo
    g1.s7 = 0u;
    asm volatile("tensor_load_to_lds %0, %1" :: "s"(g0), "s"(g1) : "memory");
}

// ---------------- Kernel 1: zero the global accumulators ----------------
__global__ void sp_zero_kernel(float* p, int n) {
    int t = blockIdx.x * blockDim.x + threadIdx.x;
    if (t < n) p[t] = 0.0f;
}

// ---------------- Kernel 2: LDS-privatized scatter reduce ----------------
// grid = B * CB * SLABS, block = BSZ.
// gsum layout: [B][C][NL]  (matches LDS [c][lab] rows -> coalesced flush)
// gcnt layout: [B][NL]
__global__ void sp_accum_kernel(const float* __restrict__ x,
                                const int*   __restrict__ labels,
                                float* __restrict__ gsum,
                                float* __restrict__ gcnt) {
    extern __shared__ float smem[];
    float* lsum = smem;                       // CBS*NL
    float* lcnt = smem + CBS * NL;            // NL
    int*   lbuf = (int*)(smem + LDS_FLOATS);  // PSLAB labels, 16B aligned

    const int tid  = threadIdx.x;
    const int wg   = blockIdx.x;
    const int slab = wg % SLABS;
    const int cb   = (wg / SLABS) % CB;
    const int b    = wg / (SLABS * CB);

    const int    pix0    = slab * PSLAB;
    const int*   labbase = labels + (size_t)b * HW + pix0;
    const float* xbase   = x + ((size_t)b * Cq + (size_t)cb * CBS) * HW + pix0;

    // Wave 0 issues one TDM descriptor: 64 KB of labels -> LDS (TENSORcnt).
    if (tid < 32) {
        tdm_load_to_lds(labbase, lbuf, PSLAB);
        wait_tensorcnt0();
    }

    for (int i = tid; i < CBS * NL; i += BSZ) lsum[i] = 0.0f;
    if (cb == 0)
        for (int i = tid; i < NL; i += BSZ) lcnt[i] = 0.0f;
    __syncthreads();   // labels staged + accumulators zeroed

    constexpr int ITER = PSLAB / (4 * BSZ);   // 8
    for (int it = 0; it < ITER; ++it) {
        const int  base = it * 4 * BSZ + tid * 4;     // 4 pixels per thread
        const int4 lv   = *(const int4*)&lbuf[base];  // one ds_load_b128

        #pragma unroll 4
        for (int c = 0; c < CBS; ++c) {
            const float4 xv = *(const float4*)(xbase + (size_t)c * HW + base);
            float* row = lsum + c * NL;   // bank = (c*NL+lab)%64 -> randomized over lanes
            lds_atomic_add(row + lv.x, xv.x);
            lds_atomic_add(row + lv.y, xv.y);
            lds_atomic_add(row + lv.z, xv.z);
            lds_atomic_add(row + lv.w, xv.w);
        }
        if (cb == 0) {   // counts: once per pixel per image
            lds_atomic_add(lcnt + lv.x, 1.0f);
            lds_atomic_add(lcnt + lv.y, 1.0f);
            lds_atomic_add(lcnt + lv.z, 1.0f);
            lds_atomic_add(lcnt + lv.w, 1.0f);
        }
    }
    __syncthreads();

    // Flush privatized accumulators with agent-scope fp32 atomics.
    float* gs = gsum + ((size_t)b * Cq + (size_t)cb * CBS) * NL;
    for (int i = tid; i < CBS * NL; i += BSZ) glb_atomic_add(gs + i, lsum[i]);
    if (cb == 0) {
        float* gc = gcnt + (size_t)b * NL;
        for (int i = tid; i < NL; i += BSZ) glb_atomic_add(gc + i, lcnt[i]);
    }
}

// ---------------- Kernel 3: means  [B][C][NL] -> [B][NL][C] with divide ----------------
__global__ void sp_means_kernel(const float* __restrict__ gsum,
                                const float* __restrict__ gcnt,
                                float* __restrict__ means) {
    int t = blockIdx.x * blockDim.x + threadIdx.x;   // lab fastest -> coalesced reads
    if (t >= Bq * Cq * NL) return;
    int lab = t % NL;
    int c   = (t / NL) % Cq;
    int b   = t / (NL * Cq);
    float cnt = gcnt[(size_t)b * NL + lab];
    float m   = gsum[t] / fmaxf(cnt, 1.0f);
    means[((size_t)b * NL + lab) * Cq + c] = m;
}

// ---------------- Kernel 4: edge gather + y ----------------
// grid = B * (NS/ET), block = 256. lane = channel -> coalesced means reads & out writes.
// out = [ xa (B*NS*C) | xb (B*NS*C) | y (B*NS) ]
__global__ void sp_gather_kernel(const float* __restrict__ means,
                                 const int*   __restrict__ edges,
                                 const float* __restrict__ ew,
                                 float* __restrict__ out) {
    extern __shared__ int ebuf[];   // ET*2 endpoint labels (2 KB)

    const int tid  = threadIdx.x;
    const int tile = blockIdx.x % (NS / ET);
    const int b    = blockIdx.x / (NS / ET);
    const int e0   = tile * ET;

    // Stage the edge tile into LDS: 256 lanes x 8 B async copies (ASYNCcnt).
    async_copy_b64(edges + ((size_t)b * NS + e0) * 2 + tid * 2, &ebuf[tid * 2]);
    wait_asynccnt0();
    __syncthreads();

    const int c  = tid & (Cq - 1);   // 0..63
    const int er = tid >> 6;         // 0..3
    constexpr size_t XB_OFF = (size_t)Bq * NS * Cq;
    constexpr size_t Y_OFF  = 2 * XB_OFF;

    #pragma unroll 4
    for (int k = 0; k < ET / 4; ++k) {
        const int e  = er + k * 4;          // local edge index (same across a wave)
        const int la = ebuf[2 * e];         // LDS broadcast reads
        const int lb = ebuf[2 * e + 1];
        const size_t o = ((size_t)b * NS + e0 + e) * Cq + c;
        out[o]          = means[((size_t)b * NL + la) * Cq + c];   // coalesced 128B/wave
        out[XB_OFF + o] = means[((size_t)b * NL + lb) * Cq + c];
    }
    // y: one coalesced copy per tile.
    out[Y_OFF + (size_t)b * NS + e0 + tid] = ew[(size_t)b * NS + e0 + tid];
}

extern "C" void kernel_launch(void* const* d_in, const int* in_sizes, int n_in,
                              void* d_out, int out_size, void* d_ws, size_t ws_size,
                              hipStream_t stream) {
    const float* x      = (const float*)d_in[0];
    const int*   labels = (const int*)d_in[1];
    const int*   edges  = (const int*)d_in[2];
    const float* ew     = (const float*)d_in[3];
    // d_in[4] = n_labels (compile-time NL here)
    float* out = (float*)d_out;

    // Workspace: gsum [B][C][NL] | gcnt [B][NL] | means [B][NL][C]  (~2.06 MB)
    float* gsum  = (float*)d_ws;
    float* gcnt  = gsum + (size_t)Bq * Cq * NL;
    float* means = gcnt + (size_t)Bq * NL;

    const int ZN = Bq * Cq * NL + Bq * NL;
    sp_zero_kernel<<<(ZN + 255) / 256, 256, 0, stream>>>(gsum, ZN);

    sp_accum_kernel<<<Bq * CB * SLABS, BSZ, ACC_LDS_BYTES, stream>>>(x, labels, gsum, gcnt);

    const int MN = Bq * Cq * NL;
    sp_means_kernel<<<(MN + 255) / 256, 256, 0, stream>>>(gsum, gcnt, means);

    sp_gather_kernel<<<Bq * (NS / ET), ET, GAT_LDS_BYTES, stream>>>(means, edges, ew, out);
}